// GraphConv_6648609374330
// MI455X (gfx1250) — compile-verified
//
#include <hip/hip_runtime.h>

typedef __attribute__((ext_vector_type(2))) float v2f;
typedef __attribute__((ext_vector_type(8))) float v8f;

#define N_NODES 100000
#define N_EDGES 500000
#define D 128                 // D_IN == D_OUT == 128
#define NODE_TILES (N_NODES / 16)   // 6250, exact

// ---------------------------------------------------------------------------
// Phase 0: zero the aggregation buffer (d_ws is poisoned to 0xAA by harness)
// ---------------------------------------------------------------------------
__global__ void zero_agg_kernel(float4* __restrict__ p, int n4) {
    int i = blockIdx.x * blockDim.x + threadIdx.x;
    if (i < n4) p[i] = make_float4(0.f, 0.f, 0.f, 0.f);
}

// ---------------------------------------------------------------------------
// Phase 1: scatter-add  agg[dst] += feat[src]
// 32 threads per edge, each thread moves 4 contiguous floats (float4 gather,
// 4x global_atomic_add_f32 scatter). Atomics land in L2 (agg is 51MB << 192MB).
// ---------------------------------------------------------------------------
__global__ void scatter_add_kernel(const float* __restrict__ feat,
                                   const int*   __restrict__ esrc,
                                   const int*   __restrict__ edst,
                                   float*       __restrict__ agg) {
    long long t = (long long)blockIdx.x * blockDim.x + threadIdx.x;
    int e = (int)(t >> 5);          // edge index
    int c = ((int)t & 31) * 4;      // feature chunk base
    if (e >= N_EDGES) return;
    int s = esrc[e];
    int d = edst[e];
    float4 v = *(const float4*)(feat + (long long)s * D + c);
    float* o = agg + (long long)d * D + c;
    atomicAdd(o + 0, v.x);
    atomicAdd(o + 1, v.y);
    atomicAdd(o + 2, v.z);
    atomicAdd(o + 3, v.w);
}

// ---------------------------------------------------------------------------
// Phase 2: out = agg @ W.T + b  via V_WMMA_F32_16X16X4_F32 (full f32 path).
//
// One wave32 computes a 16-node x 128-out strip:
//   A (16x4 f32):  lane l holds row m=l%16, K-pair 2*(l/16)  -> float2 load
//   B (4x16 f32):  lane l holds col n=l%16, K-pair 2*(l/16); B(k,n)=W[n][k]
//                  so row-major W rows feed B columns directly -> float2 load
//   C/D (16x16):   vgpr r, lane l -> (m = r + 8*(l/16), n = l%16)
// Bias folded into the accumulator init (C column == l%16 for every vgpr).
//
// Per K-step: issue A + all 8 B-fragment loads into DISTINCT registers first,
// then 8 back-to-back WMMAs, so waits are staggered instead of load->wait(0)
// ->wmma serialization.
// ---------------------------------------------------------------------------
__global__ void gemm_wmma_kernel(const float* __restrict__ agg,
                                 const float* __restrict__ W,
                                 const float* __restrict__ bias,
                                 float*       __restrict__ out) {
    const int lane = threadIdx.x & 31;
    const int wave = threadIdx.x >> 5;
    const int tile = blockIdx.x * (blockDim.x >> 5) + wave;  // node tile id
    if (tile >= NODE_TILES) return;                           // wave-uniform

    const int m   = lane & 15;        // A row within tile / B column
    const int kHi = (lane >> 4) * 2;  // K-pair offset for this half-wave

    const float* aPtr = agg + ((long long)(tile * 16 + m)) * D + kHi;
    const float* wPtr = W   + (long long)m * D + kHi;  // + t*16*D per out-tile

    // Init accumulators with the bias (broadcast across the 8 row-VGPRs).
    v8f acc[8];
#pragma unroll
    for (int t = 0; t < 8; ++t) {
        float bv = bias[t * 16 + m];
        acc[t] = (v8f){bv, bv, bv, bv, bv, bv, bv, bv};
    }

    // K loop: 32 steps of K=4; A frag reused across all 8 output tiles.
#pragma unroll 4
    for (int kk = 0; kk < D; kk += 4) {
        // ---- load phase: 9 independent b64 loads in flight ----
        v2f a = *(const v2f*)(aPtr + kk);
        v2f bf[8];
#pragma unroll
        for (int t = 0; t < 8; ++t)
            bf[t] = *(const v2f*)(wPtr + t * 16 * D + kk);
        // ---- compute phase: 8 back-to-back WMMAs ----
#pragma unroll
        for (int t = 0; t < 8; ++t)
            acc[t] = __builtin_amdgcn_wmma_f32_16x16x4_f32(
                /*neg_a=*/false, a, /*neg_b=*/false, bf[t],
                /*c_mod=*/(short)0, acc[t],
                /*reuse_a=*/false, /*reuse_b=*/false);
    }

    // Store: row = r + 8*(lane/16), col = t*16 + lane%16. Lanes 0-15 cover 16
    // consecutive columns -> 64B contiguous segments per store.
    float* oBase = out + (long long)(tile * 16) * D;
    const int rowHi = 8 * (lane >> 4);
#pragma unroll
    for (int t = 0; t < 8; ++t) {
#pragma unroll
        for (int r = 0; r < 8; ++r) {
            oBase[(long long)(r + rowHi) * D + (t * 16 + m)] = acc[t][r];
        }
    }
}

// ---------------------------------------------------------------------------
extern "C" void kernel_launch(void* const* d_in, const int* in_sizes, int n_in,
                              void* d_out, int out_size, void* d_ws, size_t ws_size,
                              hipStream_t stream) {
    const float* feat = (const float*)d_in[0];
    const int*   esrc = (const int*)d_in[1];
    const int*   edst = (const int*)d_in[2];
    const float* W    = (const float*)d_in[3];
    const float* bias = (const float*)d_in[4];
    float*       out  = (float*)d_out;
    float*       agg  = (float*)d_ws;   // 100000*128*4 = 51.2 MB scratch

    (void)in_sizes; (void)n_in; (void)out_size; (void)ws_size;

    // Phase 0: zero agg
    int n4 = N_NODES * D / 4;
    zero_agg_kernel<<<(n4 + 255) / 256, 256, 0, stream>>>((float4*)agg, n4);

    // Phase 1: scatter-add (32 threads/edge)
    long long nthreads = (long long)N_EDGES * 32;
    scatter_add_kernel<<<(unsigned)((nthreads + 255) / 256), 256, 0, stream>>>(
        feat, esrc, edst, agg);

    // Phase 2: WMMA GEMM + bias (8 waves/block, 1 node-tile per wave)
    gemm_wmma_kernel<<<(NODE_TILES + 7) / 8, 256, 0, stream>>>(agg, W, bias, out);
}